// ProposalLayer_47665547051879
// MI455X (gfx1250) — compile-verified
//
#include <hip/hip_runtime.h>
#include <hip/hip_bf16.h>
#include <stdint.h>

// ---------------------------------------------------------------------------
// ProposalLayer for MI455X (gfx1250, wave32).
//   Kernel 1: per-batch exact top-6000 (14-bit radix histogram select in LDS +
//             in-place bitonic sort of <=8192 candidates, 64KB dynamic LDS).
//             Suffix scan is wave32-shuffle based (serial part = 32 waves).
//   Kernel 2: per-batch greedy NMS; sorted index table staged into LDS with
//             gfx1250 async global->LDS B128 copies (ASYNCcnt); boxes live in
//             registers (6 slots/thread); wave32 shuffle min-reductions.
// Workspace use: B * 6000 * 4 bytes (sorted top-k indices).
// ---------------------------------------------------------------------------

#define NUM_A   9
#define K_PRE   6000
#define POST    300
#define NMS_T   0.7f
#define NBKT    16384      // 14-bit buckets
#define SORTN   8192
#define TPB     1024
#define SLOTS   6          // 6*1024 >= 6000

// generate_anchors(16, [0.5,1,2], [8,16,32]) evaluated exactly (numpy
// round-half-to-even): ratio row-major, then scale.
__device__ __constant__ float c_anchors[NUM_A][4] = {
    { -84.f,  -40.f,  99.f,  55.f}, {-176.f,  -88.f, 191.f, 103.f},
    {-360.f, -184.f, 375.f, 199.f}, { -56.f,  -56.f,  71.f,  71.f},
    {-120.f, -120.f, 135.f, 135.f}, {-248.f, -248.f, 263.f, 263.f},
    { -36.f,  -80.f,  51.f,  95.f}, { -80.f, -168.f,  95.f, 183.f},
    {-168.f, -344.f, 183.f, 359.f}
};

// Monotonic float -> uint32 key (ascending order preserved).
__device__ __forceinline__ uint32_t fkey(float f) {
    uint32_t u = __float_as_uint(f);
    return (u & 0x80000000u) ? ~u : (u | 0x80000000u);
}

// ---------------------------------------------------------------------------
// Kernel 1: exact sorted top-K per batch.  One 1024-thread workgroup per batch.
// Dynamic LDS (64KB) is first a 16384-entry histogram, then reused as the
// 8192-entry u64 bitonic sort buffer.
// ---------------------------------------------------------------------------
extern "C" __global__ void __launch_bounds__(TPB)
proposal_topk_kernel(const float* __restrict__ cls,
                     const int* __restrict__ pH, const int* __restrict__ pW,
                     uint32_t* __restrict__ topIdx) {
    const int H = *pH, W = *pW;
    const int HW = H * W;
    const int b    = blockIdx.x;
    const int tid  = threadIdx.x;
    const int lane = tid & 31, wv = tid >> 5;

    extern __shared__ unsigned char smemRaw[];
    uint32_t* hist = (uint32_t*)smemRaw;      // [NBKT]   phase 1
    uint64_t* buf  = (uint64_t*)smemRaw;      // [SORTN]  phase 2 (reuse)

    __shared__ uint32_t waveTot[TPB / 32];
    __shared__ uint32_t waveSuf[TPB / 32];
    __shared__ uint32_t sT, sChi, cntHi, cntTie;

    const float* base = cls + (size_t)(b * 2 * NUM_A + NUM_A) * HW; // skip bg

    // ---- phase 1: histogram of top-14 key bits --------------------------
    if (tid == 0) { sT = 0u; sChi = 0u; }
    for (int i = tid; i < NBKT; i += TPB) hist[i] = 0u;
    __syncthreads();

    for (int c = 0; c < NUM_A; ++c) {
        const float* ch = base + (size_t)c * HW;
        for (int p = tid; p < HW; p += TPB)
            atomicAdd(&hist[fkey(ch[p]) >> 18], 1u);
    }
    __syncthreads();

    // ---- find threshold bucket T: suffix(T) >= K, suffix(T+1) < K -------
    const int CH = NBKT / TPB;                 // 16 consecutive buckets/thread
    const int b0 = tid * CH;
    uint32_t cs = 0;
    for (int i = 0; i < CH; ++i) cs += hist[b0 + i];

    // wave32 inclusive suffix sum over lanes (sum over lanes >= lane)
    uint32_t acc = cs;
    #pragma unroll
    for (int m = 1; m < 32; m <<= 1) {
        uint32_t t = (uint32_t)__shfl_down((int)acc, m, 32);
        if (lane + m < 32) acc += t;
    }
    if (lane == 0) waveTot[wv] = acc;          // wave total
    __syncthreads();
    if (tid == 0) {                            // serial part: 32 entries only
        uint32_t run = 0;
        for (int w2 = (TPB / 32) - 1; w2 >= 0; --w2) { waveSuf[w2] = run; run += waveTot[w2]; }
    }
    __syncthreads();
    const uint32_t mySufHi = waveSuf[wv] + (acc - cs);  // strictly higher chunks

    if (mySufHi < K_PRE && mySufHi + cs >= K_PRE) {
        uint32_t run = mySufHi;                // unique crossing thread
        for (int i = CH - 1; i >= 0; --i) {
            uint32_t bk = (uint32_t)(b0 + i);
            uint32_t nr = run + hist[bk];
            if (nr >= K_PRE) { sT = bk; sChi = run; break; }
            run = nr;
        }
    }
    __syncthreads();
    const uint32_t T = sT, Chi = sChi;
    if (tid == 0) { cntHi = 0u; cntTie = Chi; }
    __syncthreads();   // all hist reads done; buf may now alias hist

    // ---- phase 2: compact candidates into sort buffer -------------------
    // key64 = (score_key << 32) | ~idx  -> desc sort == score desc, idx asc
    for (int c = 0; c < NUM_A; ++c) {
        const float* ch = base + (size_t)c * HW;
        for (int p = tid; p < HW; p += TPB) {
            uint32_t k  = fkey(ch[p]);
            uint32_t bk = k >> 18;
            if (bk > T) {
                uint32_t pos = atomicAdd(&cntHi, 1u);
                if (pos < SORTN) {
                    uint32_t idx = (uint32_t)(p * NUM_A + c);
                    buf[pos] = ((uint64_t)k << 32) | (uint32_t)(~idx);
                }
            } else if (bk == T) {
                uint32_t pos = atomicAdd(&cntTie, 1u);
                if (pos < SORTN) {
                    uint32_t idx = (uint32_t)(p * NUM_A + c);
                    buf[pos] = ((uint64_t)k << 32) | (uint32_t)(~idx);
                }
            }
        }
    }
    __syncthreads();
    uint32_t total = cntTie; if (total > SORTN) total = SORTN;
    for (uint32_t i = total + tid; i < SORTN; i += TPB) buf[i] = 0ull; // pad sinks
    __syncthreads();

    // ---- bitonic sort, descending, 8192 x u64 in LDS --------------------
    for (uint32_t k = 2; k <= SORTN; k <<= 1) {
        for (uint32_t j = k >> 1; j > 0; j >>= 1) {
            for (uint32_t i = tid; i < SORTN; i += TPB) {
                uint32_t l = i ^ j;
                if (l > i) {
                    uint64_t x = buf[i], y = buf[l];
                    bool up = ((i & k) == 0);           // descending network
                    if (up ? (x < y) : (x > y)) { buf[i] = y; buf[l] = x; }
                }
            }
            __syncthreads();
        }
    }

    for (int j = tid; j < K_PRE; j += TPB)
        topIdx[(size_t)b * K_PRE + j] = ~(uint32_t)(buf[j] & 0xFFFFFFFFull);
}

// ---------------------------------------------------------------------------
// Kernel 2: greedy NMS.  One 1024-thread workgroup per batch.
// ---------------------------------------------------------------------------
extern "C" __global__ void __launch_bounds__(TPB)
proposal_nms_kernel(const float* __restrict__ deltas,
                    const float* __restrict__ im_shape,
                    const uint32_t* __restrict__ topIdx,
                    float* __restrict__ out,
                    const int* __restrict__ pH, const int* __restrict__ pW) {
    const int H = *pH, W = *pW;
    const int HW = H * W;
    const int b   = blockIdx.x;
    const int tid = threadIdx.x;

    __shared__ __align__(16) uint32_t sIdx[SLOTS * TPB];
    __shared__ uint32_t red[TPB / 32];
    __shared__ uint32_t sWin;
    __shared__ float    bc[5];

    // ---- stage sorted index row into LDS via gfx1250 async B128 copies --
    // K_PRE u32 == K_PRE/4 x 16-byte transfers (LDS dest 16B-aligned).
    const uint32_t* src = topIdx + (size_t)b * K_PRE;
    for (int t = tid; t < K_PRE / 4; t += TPB) {
        uint32_t ldsOff = (uint32_t)(uintptr_t)(&sIdx[t * 4]); // low 32b = LDS addr
        uint32_t gOff   = (uint32_t)t * 16u;
        asm volatile("global_load_async_to_lds_b128 %0, %1, %2"
                     :: "v"(ldsOff), "v"(gOff), "s"(src) : "memory");
    }
    for (int j = K_PRE + tid; j < SLOTS * TPB; j += TPB) sIdx[j] = 0xFFFFFFFFu;
    asm volatile("s_wait_asynccnt 0" ::: "memory");
    __syncthreads();

    // ---- decode + clip the 6000 selected boxes into registers -----------
    const float ximax = im_shape[b * 2 + 1] - 1.0f;   // width  - 1
    const float yimax = im_shape[b * 2 + 0] - 1.0f;   // height - 1
    const float* db = deltas + (size_t)b * (4 * NUM_A) * HW;

    float rx1[SLOTS], ry1[SLOTS], rx2[SLOTS], ry2[SLOTS], rar[SLOTS];
    uint32_t valid = 0u;
    #pragma unroll
    for (int s = 0; s < SLOTS; ++s) {
        int j = s * TPB + tid;
        uint32_t idx = sIdx[j];
        rx1[s] = ry1[s] = rx2[s] = ry2[s] = 0.f; rar[s] = 0.f;
        if (j < K_PRE && idx < (uint32_t)(HW * NUM_A)) {
            int a = (int)(idx % NUM_A);
            int p = (int)(idx / NUM_A);
            int y = p / W, x = p - y * W;
            size_t o = (size_t)(a * 4) * HW + (size_t)y * W + x;
            float dx = db[o], dy = db[o + (size_t)HW];
            float dw = db[o + 2 * (size_t)HW], dh = db[o + 3 * (size_t)HW];
            float sx = (float)x * 16.0f, sy = (float)y * 16.0f;
            float a0 = c_anchors[a][0] + sx, a1 = c_anchors[a][1] + sy;
            float a2 = c_anchors[a][2] + sx, a3 = c_anchors[a][3] + sy;
            float w  = a2 - a0 + 1.0f,  h  = a3 - a1 + 1.0f;
            float cx = a0 + 0.5f * w,   cy = a1 + 0.5f * h;
            float pcx = dx * w + cx,    pcy = dy * h + cy;
            float pw  = expf(dw) * w,   ph  = expf(dh) * h;
            float x1 = fminf(fmaxf(pcx - 0.5f * pw, 0.f), ximax);
            float y1 = fminf(fmaxf(pcy - 0.5f * ph, 0.f), yimax);
            float x2 = fminf(fmaxf(pcx + 0.5f * pw, 0.f), ximax);
            float y2 = fminf(fmaxf(pcy + 0.5f * ph, 0.f), yimax);
            rx1[s] = x1; ry1[s] = y1; rx2[s] = x2; ry2[s] = y2;
            rar[s] = (x2 - x1 + 1.f) * (y2 - y1 + 1.f);   // area of clipped box
            valid |= (1u << s);
        }
    }

    // ---- greedy NMS: 300 sequential picks -------------------------------
    const float fb = (float)b;
    const int   wv = tid >> 5, lane = tid & 31;
    for (int it = 0; it < POST; ++it) {
        // smallest surviving global rank owned by this thread
        uint32_t cand = 0xFFFFFFFFu;
        #pragma unroll
        for (int s = SLOTS - 1; s >= 0; --s)
            if ((valid >> s) & 1u) cand = (uint32_t)(s * TPB + tid);
        // wave32 min
        #pragma unroll
        for (int m = 16; m > 0; m >>= 1) {
            uint32_t o = (uint32_t)__shfl_xor((int)cand, m, 32);
            cand = (cand < o) ? cand : o;
        }
        if (lane == 0) red[wv] = cand;
        __syncthreads();
        if (wv == 0) {
            uint32_t v = red[lane];
            #pragma unroll
            for (int m = 16; m > 0; m >>= 1) {
                uint32_t o = (uint32_t)__shfl_xor((int)v, m, 32);
                v = (v < o) ? v : o;
            }
            if (lane == 0) sWin = v;
        }
        __syncthreads();
        uint32_t win = sWin;

        if (win == 0xFFFFFFFFu) {              // nothing left: emit [b,0,0,0,0]
            if (tid == 0) {
                float* o = out + ((size_t)b * POST + it) * 5;
                o[0] = fb; o[1] = 0.f; o[2] = 0.f; o[3] = 0.f; o[4] = 0.f;
            }
            continue;                           // uniform branch, barriers stay matched
        }

        uint32_t wt = win & (TPB - 1u);
        uint32_t ws = win >> 10;
        if (tid == (int)wt) {                   // owner broadcasts + retires its box
            float sx1 = 0, sy1 = 0, sx2 = 0, sy2 = 0, sa = 0;
            #pragma unroll
            for (int s = 0; s < SLOTS; ++s)
                if (ws == (uint32_t)s) { sx1 = rx1[s]; sy1 = ry1[s];
                                         sx2 = rx2[s]; sy2 = ry2[s]; sa = rar[s]; }
            bc[0] = sx1; bc[1] = sy1; bc[2] = sx2; bc[3] = sy2; bc[4] = sa;
            valid &= ~(1u << ws);
        }
        __syncthreads();
        float wx1 = bc[0], wy1 = bc[1], wx2 = bc[2], wy2 = bc[3], wa = bc[4];
        if (tid == 0) {
            float* o = out + ((size_t)b * POST + it) * 5;
            o[0] = fb; o[1] = wx1; o[2] = wy1; o[3] = wx2; o[4] = wy2;
        }
        #pragma unroll
        for (int s = 0; s < SLOTS; ++s) {
            if ((valid >> s) & 1u) {
                float iw = fminf(rx2[s], wx2) - fmaxf(rx1[s], wx1) + 1.0f;
                float ih = fminf(ry2[s], wy2) - fmaxf(ry1[s], wy1) + 1.0f;
                iw = fmaxf(iw, 0.f); ih = fmaxf(ih, 0.f);
                float inter = iw * ih;
                float iou = inter / (rar[s] + wa - inter);
                if (iou > NMS_T) valid &= ~(1u << s);
            }
        }
        // next iteration's red/bc writes are fenced by its own barriers
    }
}

// ---------------------------------------------------------------------------
extern "C" void kernel_launch(void* const* d_in, const int* in_sizes, int n_in,
                              void* d_out, int out_size, void* d_ws, size_t ws_size,
                              hipStream_t stream) {
    const float* cls      = (const float*)d_in[0];
    const float* deltas   = (const float*)d_in[1];
    const float* im_shape = (const float*)d_in[2];
    const int*   pH       = (const int*)d_in[3];
    const int*   pW       = (const int*)d_in[4];
    float*       out      = (float*)d_out;
    uint32_t*    topIdx   = (uint32_t*)d_ws;      // B * 6000 u32

    const int B = in_sizes[2] / 2;                // im_shape is (B,2)

    // 64KB dynamic LDS for the histogram/sort buffer (gfx1250 WGP has 320KB).
    (void)hipFuncSetAttribute((const void*)proposal_topk_kernel,
                              hipFuncAttributeMaxDynamicSharedMemorySize,
                              NBKT * (int)sizeof(uint32_t));

    proposal_topk_kernel<<<B, TPB, NBKT * sizeof(uint32_t), stream>>>(
        cls, pH, pW, topIdx);
    proposal_nms_kernel<<<B, TPB, 0, stream>>>(
        deltas, im_shape, topIdx, out, pH, pW);

    (void)n_in; (void)out_size; (void)ws_size;
}